// AttentionTopologyModule_26405458936159
// MI455X (gfx1250) — compile-verified
//
#include <hip/hip_runtime.h>
#include <math.h>

#define BB 4
#define NN 8192
#define CC 64
#define KK 16
#define HH 64
#define CIN1 131      // 2C+3
#define CIN1P 132     // padded to multiple of 4
#define CINV 67       // C+3
#define CINVP 68
#define EPSB 1e-5f
#define NB2 512       // blocks for MLP stats / attn kernels
#define NBO 256       // blocks for output conv

typedef float v2f __attribute__((ext_vector_type(2)));
typedef float v8f __attribute__((ext_vector_type(8)));

static __device__ __forceinline__ v8f wmma4(v2f a, v2f b, v8f c) {
  // D = A(16x4 f32) * B(4x16 f32) + C(16x16 f32)
  return __builtin_amdgcn_wmma_f32_16x16x4_f32(false, a, false, b, (short)0, c,
                                               false, false);
}

// ---------------------------------------------------------------- transpose
// feats [B,C,N] -> featsT [B,N,C];  xyz [B,3,N] -> xyzT [B,N,3]
__global__ void k_transpose(const float* __restrict__ xyz,
                            const float* __restrict__ feats,
                            float* __restrict__ xyzT,
                            float* __restrict__ featsT) {
  int i = blockIdx.x * 256 + threadIdx.x;           // exact grid: B*N*C
  int b = i / (NN * CC);
  int rem = i - b * (NN * CC);
  int n = rem / CC;
  int c = rem - n * CC;
  featsT[i] = feats[((size_t)b * CC + c) * NN + n];
  if (i < BB * NN * 3) {
    int b2 = i / (NN * 3);
    int r2 = i - b2 * (NN * 3);
    int n2 = r2 / 3;
    int d2 = r2 - n2 * 3;
    xyzT[i] = xyz[((size_t)b2 * 3 + d2) * NN + n2];
  }
}

// ------------------------------------------------------- kNN (K=16) via WMMA
// d[m][n] = sq_m + (-2 x_m . q_n) + sq_n  as one V_WMMA_F32_16X16X4_F32 per
// 16x16 tile: A[m]=(x,y,z,sq_m), B[.][n]=(-2qx,-2qy,-2qz,1), C[m][n]=sq_n.
// Wave owns 16 queries (columns). Lane pair (r, r+16) splits the candidate
// stream; each keeps a register top-16, merged 32->16 per query at the end.
__global__ __launch_bounds__(256) void k_knn(const float* __restrict__ xyzT,
                                             int* __restrict__ idxb) {
  __shared__ float tile[256][4];   // staged candidates: x,y,z,sq
  __shared__ float mD[8][16][32];  // per-wave merge buffers
  __shared__ int mI[8][16][32];
  int tid = threadIdx.x;
  int wid = tid >> 5, lane = tid & 31, half = lane >> 4, r = lane & 15;
  int qblock = blockIdx.x * 128;     // 128 queries per block (same b)
  int b = qblock >> 13;
  int p = qblock + wid * 16 + r;     // this lane's query (global point id)
  int n = p & (NN - 1);
  const float* qp = xyzT + (size_t)(b * NN + n) * 3;
  float qx = qp[0], qy = qp[1], qz = qp[2];
  float sqq = qx * qx + qy * qy + qz * qz;
  v2f bfrag;
  if (half == 0) { bfrag.x = -2.f * qx; bfrag.y = -2.f * qy; }
  else           { bfrag.x = -2.f * qz; bfrag.y = 1.0f; }
  float bd[KK];
  int bi[KK];
#pragma unroll
  for (int i = 0; i < KK; ++i) { bd[i] = 3.0e38f; bi[i] = 0; }
  float cmax = 3.0e38f;
  int cslot = 0;
  const float* base = xyzT + (size_t)b * NN * 3;
  for (int t0 = 0; t0 < NN; t0 += 256) {
    __syncthreads();
    {
      int m = t0 + tid;
      float x = base[m * 3 + 0], y = base[m * 3 + 1], z = base[m * 3 + 2];
      tile[tid][0] = x;
      tile[tid][1] = y;
      tile[tid][2] = z;
      tile[tid][3] = x * x + y * y + z * z;
    }
    if (t0 + 256 < NN)
      __builtin_prefetch(&base[(t0 + 256 + tid) * 3], 0, 1);
    __syncthreads();
    for (int w = 0; w < 16; ++w) {   // 16 candidate sub-tiles of 16
      int lm = w * 16 + r;
      v2f a;
      a.x = tile[lm][2 * half];
      a.y = tile[lm][2 * half + 1];
      v8f acc;
#pragma unroll
      for (int j = 0; j < 8; ++j) acc[j] = sqq;   // C[m][n] = sq_n
      acc = wmma4(a, bfrag, acc);
#pragma unroll
      for (int j = 0; j < 8; ++j) {
        float d = acc[j];
        if (d < cmax) {   // strict < keeps earlier index on ties
          int m = t0 + w * 16 + j + 8 * half;
#pragma unroll
          for (int i = 0; i < KK; ++i)
            if (i == cslot) { bd[i] = d; bi[i] = m; }
          cmax = bd[0]; cslot = 0;
#pragma unroll
          for (int i = 1; i < KK; ++i)
            if (bd[i] > cmax) { cmax = bd[i]; cslot = i; }
        }
      }
    }
  }
  // merge lane pair's top-16 lists -> final top-16 per query
#pragma unroll
  for (int k = 0; k < KK; ++k) {
    mD[wid][r][half * 16 + k] = bd[k];
    mI[wid][r][half * 16 + k] = bi[k];
  }
  __syncthreads();
  if (half == 0) {
    for (int k = 0; k < KK; ++k) {
      float best = mD[wid][r][0];
      int bidx = mI[wid][r][0];
      int bslot = 0;
      for (int s = 1; s < 32; ++s) {
        float dv = mD[wid][r][s];
        int iv = mI[wid][r][s];
        if (dv < best || (dv == best && iv < bidx)) {
          best = dv; bidx = iv; bslot = s;
        }
      }
      idxb[(size_t)p * KK + k] = bidx;
      mD[wid][r][bslot] = 3.0e38f;
    }
  }
}

// Build the 16 concatenated rows [ctr(64) | nbr(64) | rel_xyz(3) | pad] in LDS.
static __device__ __forceinline__ void fill_rows(
    float (*rows)[CIN1P], const float* __restrict__ featsT,
    const float* __restrict__ xyzT, const int* __restrict__ idxb, int b, int n,
    int p, int lane) {
  const float* ctr = featsT + (size_t)(b * NN + n) * CC;
  for (int rr = 0; rr < KK; ++rr) {
    int m = idxb[(size_t)p * KK + rr];
    const float* nb = featsT + (size_t)(b * NN + m) * CC;
    rows[rr][lane] = ctr[lane];
    rows[rr][lane + 32] = ctr[lane + 32];
    rows[rr][lane + 64] = nb[lane];
    rows[rr][lane + 96] = nb[lane + 32];
    if (lane < 4) {
      float v = 0.f;
      if (lane < 3)
        v = xyzT[(size_t)(b * NN + m) * 3 + lane] -
            xyzT[(size_t)(b * NN + n) * 3 + lane];
      rows[rr][128 + lane] = v;   // col 131 zero pad
    }
  }
}

// --------------------------------------------- pass 1: h/v BN statistics
__global__ __launch_bounds__(256) void k_mlp_stats(
    const float* __restrict__ featsT, const float* __restrict__ xyzT,
    const int* __restrict__ idxb, const float* __restrict__ W1,
    const float* __restrict__ b1, const float* __restrict__ Wv,
    const float* __restrict__ bv, float* __restrict__ partH,
    float* __restrict__ partV) {
  __shared__ float Ws1[CIN1P][HH];
  __shared__ float Wvs[CINVP][HH];
  __shared__ float rowsBuf[8][KK][CIN1P];
  __shared__ float red[8][2][HH][2];
  int tid = threadIdx.x;
  for (int i = tid; i < CIN1P * HH; i += 256) {
    int c = i / HH, h = i % HH;
    Ws1[c][h] = (c < CIN1) ? W1[h * CIN1 + c] : 0.f;
  }
  for (int i = tid; i < CINVP * HH; i += 256) {
    int c = i / HH, h = i % HH;
    Wvs[c][h] = (c < CINV) ? Wv[h * CINV + c] : 0.f;
  }
  __syncthreads();
  int wid = tid >> 5, lane = tid & 31, half = lane >> 4, r = lane & 15;
  float sH[4] = {0.f, 0.f, 0.f, 0.f}, qH[4] = {0.f, 0.f, 0.f, 0.f};
  float sV[4] = {0.f, 0.f, 0.f, 0.f}, qV[4] = {0.f, 0.f, 0.f, 0.f};
  const int TOT = BB * NN;       // 32768 tiles, NW=4096 -> uniform 8 iters/wave
  const int NW = NB2 * 8;
  v8f z = {0.f, 0.f, 0.f, 0.f, 0.f, 0.f, 0.f, 0.f};
  for (int t = blockIdx.x * 8 + wid; t < TOT; t += NW) {
    int b = t >> 13, n = t & (NN - 1);
    __syncthreads();
    fill_rows(rowsBuf[wid], featsT, xyzT, idxb, b, n, t, lane);
    __syncthreads();
    v8f accH[4], accV[4];
#pragma unroll
    for (int nt = 0; nt < 4; ++nt) { accH[nt] = z; accV[nt] = z; }
    for (int kk = 0; kk < 33; ++kk) {           // attn branch: 132 = 33*4
      int c0 = 4 * kk + 2 * half;
      v2f a;
      a.x = rowsBuf[wid][r][c0];
      a.y = rowsBuf[wid][r][c0 + 1];
#pragma unroll
      for (int nt = 0; nt < 4; ++nt) {
        v2f bb;
        bb.x = Ws1[c0][nt * 16 + r];
        bb.y = Ws1[c0 + 1][nt * 16 + r];
        accH[nt] = wmma4(a, bb, accH[nt]);
      }
    }
    for (int kk = 0; kk < 17; ++kk) {           // value branch: 68 = 17*4
      int cw = 4 * kk + 2 * half;
      int c0 = 64 + cw;
      v2f a;
      a.x = rowsBuf[wid][r][c0];
      a.y = rowsBuf[wid][r][c0 + 1];
#pragma unroll
      for (int nt = 0; nt < 4; ++nt) {
        v2f bb;
        bb.x = Wvs[cw][nt * 16 + r];
        bb.y = Wvs[cw + 1][nt * 16 + r];
        accV[nt] = wmma4(a, bb, accV[nt]);
      }
    }
#pragma unroll
    for (int nt = 0; nt < 4; ++nt) {
      int ch = nt * 16 + r;
      float bb1 = b1[ch], bbv = bv[ch];
#pragma unroll
      for (int j = 0; j < 8; ++j) {
        float hv = accH[nt][j] + bb1;
        sH[nt] += hv; qH[nt] += hv * hv;
        float vv = accV[nt][j] + bbv;
        sV[nt] += vv; qV[nt] += vv * vv;
      }
    }
  }
#pragma unroll
  for (int nt = 0; nt < 4; ++nt) {
    sH[nt] += __shfl_xor(sH[nt], 16);
    qH[nt] += __shfl_xor(qH[nt], 16);
    sV[nt] += __shfl_xor(sV[nt], 16);
    qV[nt] += __shfl_xor(qV[nt], 16);
  }
  if (half == 0) {
#pragma unroll
    for (int nt = 0; nt < 4; ++nt) {
      int ch = nt * 16 + r;
      red[wid][0][ch][0] = sH[nt]; red[wid][0][ch][1] = qH[nt];
      red[wid][1][ch][0] = sV[nt]; red[wid][1][ch][1] = qV[nt];
    }
  }
  __syncthreads();
  if (tid < 128) {
    int which = tid >> 6, ch = tid & 63;
    float s = 0.f, qq = 0.f;
    for (int w = 0; w < 8; ++w) { s += red[w][which][ch][0]; qq += red[w][which][ch][1]; }
    float* dst = which ? partV : partH;
    dst[(size_t)blockIdx.x * HH * 2 + ch * 2] = s;
    dst[(size_t)blockIdx.x * HH * 2 + ch * 2 + 1] = qq;
  }
}

// ---------------------------------------------- finalize BN: fold scale/shift
__global__ void k_reduce_stats(const float* __restrict__ part, int nblk,
                               const float* __restrict__ gamma,
                               const float* __restrict__ beta, float cnt,
                               float* __restrict__ fold) {
  int ch = threadIdx.x;  // 64 threads
  float s = 0.f, q = 0.f;
  for (int i = 0; i < nblk; ++i) {
    s += part[(size_t)i * 128 + ch * 2];
    q += part[(size_t)i * 128 + ch * 2 + 1];
  }
  float mean = s / cnt;
  float var = fmaxf(q / cnt - mean * mean, 0.f);
  float sc = gamma[ch] * rsqrtf(var + EPSB);
  fold[ch * 2] = sc;
  fold[ch * 2 + 1] = beta[ch] - mean * sc;
}

// --------------------- pass 2: recompute h/v, BN+relu, softmax, weighted sum
__global__ __launch_bounds__(256) void k_attn_value(
    const float* __restrict__ featsT, const float* __restrict__ xyzT,
    const int* __restrict__ idxb, const float* __restrict__ W1,
    const float* __restrict__ b1, const float* __restrict__ Wv,
    const float* __restrict__ bv, const float* __restrict__ W2,
    const float* __restrict__ b2, const float* __restrict__ foldH,
    const float* __restrict__ foldV, float* __restrict__ outfeat) {
  __shared__ float Ws1[CIN1P][HH];
  __shared__ float Wvs[CINVP][HH];
  __shared__ float rowsBuf[8][KK][CIN1P];
  __shared__ float W2s[HH];
  int tid = threadIdx.x;
  for (int i = tid; i < CIN1P * HH; i += 256) {
    int c = i / HH, h = i % HH;
    Ws1[c][h] = (c < CIN1) ? W1[h * CIN1 + c] : 0.f;
  }
  for (int i = tid; i < CINVP * HH; i += 256) {
    int c = i / HH, h = i % HH;
    Wvs[c][h] = (c < CINV) ? Wv[h * CINV + c] : 0.f;
  }
  if (tid < HH) W2s[tid] = W2[tid];
  __syncthreads();
  int wid = tid >> 5, lane = tid & 31, half = lane >> 4, r = lane & 15;
  const int TOT = BB * NN;
  const int NW = NB2 * 8;
  v8f z = {0.f, 0.f, 0.f, 0.f, 0.f, 0.f, 0.f, 0.f};
  for (int t = blockIdx.x * 8 + wid; t < TOT; t += NW) {   // uniform 8 iters
    int b = t >> 13, n = t & (NN - 1);
    __syncthreads();
    fill_rows(rowsBuf[wid], featsT, xyzT, idxb, b, n, t, lane);
    __syncthreads();
    v8f accH[4], accV[4];
#pragma unroll
    for (int nt = 0; nt < 4; ++nt) { accH[nt] = z; accV[nt] = z; }
    for (int kk = 0; kk < 33; ++kk) {
      int c0 = 4 * kk + 2 * half;
      v2f a;
      a.x = rowsBuf[wid][r][c0];
      a.y = rowsBuf[wid][r][c0 + 1];
#pragma unroll
      for (int nt = 0; nt < 4; ++nt) {
        v2f bb;
        bb.x = Ws1[c0][nt * 16 + r];
        bb.y = Ws1[c0 + 1][nt * 16 + r];
        accH[nt] = wmma4(a, bb, accH[nt]);
      }
    }
    for (int kk = 0; kk < 17; ++kk) {
      int cw = 4 * kk + 2 * half;
      int c0 = 64 + cw;
      v2f a;
      a.x = rowsBuf[wid][r][c0];
      a.y = rowsBuf[wid][r][c0 + 1];
#pragma unroll
      for (int nt = 0; nt < 4; ++nt) {
        v2f bb;
        bb.x = Wvs[cw][nt * 16 + r];
        bb.y = Wvs[cw + 1][nt * 16 + r];
        accV[nt] = wmma4(a, bb, accV[nt]);
      }
    }
    // BN fold + relu; stash h at cols [0,64) and v at cols [64,128) of rowsBuf
#pragma unroll
    for (int nt = 0; nt < 4; ++nt) {
      int ch = nt * 16 + r;
      float scH = foldH[ch * 2], shH = foldH[ch * 2 + 1] + foldH[ch * 2] * b1[ch];
      float scV = foldV[ch * 2], shV = foldV[ch * 2 + 1] + foldV[ch * 2] * bv[ch];
#pragma unroll
      for (int j = 0; j < 8; ++j) {
        int row = j + 8 * half;
        rowsBuf[wid][row][ch] = fmaxf(fmaf(scH, accH[nt][j], shH), 0.f);
        rowsBuf[wid][row][64 + ch] = fmaxf(fmaf(scV, accV[nt][j], shV), 0.f);
      }
    }
    __syncthreads();
    // logits + softmax over the 16 neighbors (rows duplicated across halves)
    int rowL = lane & 15;
    float lg = b2[0];
#pragma unroll
    for (int c = 0; c < HH; ++c) lg = fmaf(rowsBuf[wid][rowL][c], W2s[c], lg);
    float mx = lg;
#pragma unroll
    for (int off = 8; off; off >>= 1) mx = fmaxf(mx, __shfl_xor(mx, off, 16));
    float e = expf(lg - mx);
    float ssum = e;
#pragma unroll
    for (int off = 8; off; off >>= 1) ssum += __shfl_xor(ssum, off, 16);
    float attn = e / ssum;
    // out[c] = sum_k attn[k] * v[k][c]; each lane covers channels lane, lane+32
    float acc1 = 0.f, acc2 = 0.f;
#pragma unroll
    for (int rr = 0; rr < KK; ++rr) {
      float a = __shfl(attn, rr);
      acc1 = fmaf(a, rowsBuf[wid][rr][64 + lane], acc1);
      acc2 = fmaf(a, rowsBuf[wid][rr][96 + lane], acc2);
    }
    outfeat[(size_t)t * CC + lane] = acc1;
    outfeat[(size_t)t * CC + 32 + lane] = acc2;
  }
}

// ------------------------------ output 1x1 conv (WMMA) + its BN statistics
__global__ __launch_bounds__(256) void k_outconv(
    const float* __restrict__ outfeat, const float* __restrict__ Wo,
    const float* __restrict__ bo, float* __restrict__ opre,
    float* __restrict__ partO) {
  __shared__ float WoS[CC][CC];
  __shared__ float rows2[8][KK][68];
  __shared__ float red[8][CC][2];
  int tid = threadIdx.x;
  for (int i = tid; i < CC * CC; i += 256) {
    int c = i / CC, d = i % CC;
    WoS[c][d] = Wo[d * CC + c];
  }
  __syncthreads();
  int wid = tid >> 5, lane = tid & 31, half = lane >> 4, r = lane & 15;
  int t = blockIdx.x * 8 + wid;  // exactly 2048 tiles over B*N rows
  int p0 = t * KK;
  for (int rr = 0; rr < KK; ++rr) {
    rows2[wid][rr][lane] = outfeat[(size_t)(p0 + rr) * CC + lane];
    rows2[wid][rr][lane + 32] = outfeat[(size_t)(p0 + rr) * CC + 32 + lane];
  }
  __syncthreads();
  v8f z = {0.f, 0.f, 0.f, 0.f, 0.f, 0.f, 0.f, 0.f};
  v8f acc[4];
#pragma unroll
  for (int nt = 0; nt < 4; ++nt) acc[nt] = z;
  for (int kk = 0; kk < 16; ++kk) {
    int c0 = 4 * kk + 2 * half;
    v2f a;
    a.x = rows2[wid][r][c0];
    a.y = rows2[wid][r][c0 + 1];
#pragma unroll
    for (int nt = 0; nt < 4; ++nt) {
      v2f bb;
      bb.x = WoS[c0][nt * 16 + r];
      bb.y = WoS[c0 + 1][nt * 16 + r];
      acc[nt] = wmma4(a, bb, acc[nt]);
    }
  }
  float s[4] = {0.f, 0.f, 0.f, 0.f}, q[4] = {0.f, 0.f, 0.f, 0.f};
#pragma unroll
  for (int nt = 0; nt < 4; ++nt) {
    int ch = nt * 16 + r;
    float bob = bo[ch];
#pragma unroll
    for (int j = 0; j < 8; ++j) {
      int row = j + 8 * half;
      float ov = acc[nt][j] + bob;
      opre[(size_t)(p0 + row) * CC + ch] = ov;
      s[nt] += ov;
      q[nt] += ov * ov;
    }
  }
#pragma unroll
  for (int nt = 0; nt < 4; ++nt) {
    s[nt] += __shfl_xor(s[nt], 16);
    q[nt] += __shfl_xor(q[nt], 16);
  }
  if (half == 0) {
#pragma unroll
    for (int nt = 0; nt < 4; ++nt) {
      int ch = nt * 16 + r;
      red[wid][ch][0] = s[nt];
      red[wid][ch][1] = q[nt];
    }
  }
  __syncthreads();
  if (tid < CC) {
    float ss = 0.f, qq = 0.f;
    for (int w = 0; w < 8; ++w) { ss += red[w][tid][0]; qq += red[w][tid][1]; }
    partO[(size_t)blockIdx.x * CC * 2 + tid * 2] = ss;
    partO[(size_t)blockIdx.x * CC * 2 + tid * 2 + 1] = qq;
  }
}

// ------------------------------- final: BN + relu + residual + transpose out
__global__ void k_final(const float* __restrict__ opre,
                        const float* __restrict__ featsT,
                        const float* __restrict__ foldO,
                        float* __restrict__ dout) {
  int i = blockIdx.x * 256 + threadIdx.x;  // [b][c][n], exact grid
  int b = i >> 19;                         // C*N = 2^19
  int rem = i & (CC * NN - 1);
  int c = rem >> 13;
  int n = rem & (NN - 1);
  size_t src = (size_t)(b * NN + n) * CC + c;
  float v = fmaxf(fmaf(foldO[c * 2], opre[src], foldO[c * 2 + 1]), 0.f);
  dout[i] = v + featsT[src];
}

extern "C" void kernel_launch(void* const* d_in, const int* in_sizes, int n_in,
                              void* d_out, int out_size, void* d_ws,
                              size_t ws_size, hipStream_t stream) {
  const float* xyz = (const float*)d_in[0];
  const float* feats = (const float*)d_in[1];
  const float* W1 = (const float*)d_in[2];
  const float* b1 = (const float*)d_in[3];
  const float* g1 = (const float*)d_in[4];
  const float* be1 = (const float*)d_in[5];
  const float* W2 = (const float*)d_in[6];
  const float* b2 = (const float*)d_in[7];
  const float* Wv = (const float*)d_in[8];
  const float* bv = (const float*)d_in[9];
  const float* gv = (const float*)d_in[10];
  const float* bev = (const float*)d_in[11];
  const float* Wo = (const float*)d_in[12];
  const float* bo = (const float*)d_in[13];
  const float* go = (const float*)d_in[14];
  const float* beo = (const float*)d_in[15];

  char* w = (char*)d_ws;
  int* idxb = (int*)w;        w += sizeof(int) * (size_t)BB * NN * KK;
  float* featsT = (float*)w;  w += sizeof(float) * (size_t)BB * NN * CC;
  float* xyzT = (float*)w;    w += sizeof(float) * (size_t)BB * NN * 3;
  float* outfeat = (float*)w; w += sizeof(float) * (size_t)BB * NN * CC;
  float* opre = (float*)w;    w += sizeof(float) * (size_t)BB * NN * CC;
  float* partH = (float*)w;   w += sizeof(float) * (size_t)NB2 * HH * 2;
  float* partV = (float*)w;   w += sizeof(float) * (size_t)NB2 * HH * 2;
  float* partO = (float*)w;   w += sizeof(float) * (size_t)NBO * CC * 2;
  float* foldH = (float*)w;   w += sizeof(float) * 128;
  float* foldV = (float*)w;   w += sizeof(float) * 128;
  float* foldO = (float*)w;   w += sizeof(float) * 128;

  k_transpose<<<(BB * NN * CC) / 256, 256, 0, stream>>>(xyz, feats, xyzT, featsT);
  k_knn<<<(BB * NN) / 128, 256, 0, stream>>>(xyzT, idxb);
  k_mlp_stats<<<NB2, 256, 0, stream>>>(featsT, xyzT, idxb, W1, b1, Wv, bv,
                                       partH, partV);
  k_reduce_stats<<<1, 64, 0, stream>>>(partH, NB2, g1, be1,
                                       (float)((size_t)BB * NN * KK), foldH);
  k_reduce_stats<<<1, 64, 0, stream>>>(partV, NB2, gv, bev,
                                       (float)((size_t)BB * NN * KK), foldV);
  k_attn_value<<<NB2, 256, 0, stream>>>(featsT, xyzT, idxb, W1, b1, Wv, bv, W2,
                                        b2, foldH, foldV, outfeat);
  k_outconv<<<NBO, 256, 0, stream>>>(outfeat, Wo, bo, opre, partO);
  k_reduce_stats<<<1, 64, 0, stream>>>(partO, NBO, go, beo, (float)(BB * NN),
                                       foldO);
  k_final<<<(BB * CC * NN) / 256, 256, 0, stream>>>(opre, featsT, foldO,
                                                    (float*)d_out);
  (void)in_sizes; (void)n_in; (void)out_size; (void)ws_size;
}